// QLSTM_63136019251788
// MI455X (gfx1250) — compile-verified
//
#include <hip/hip_runtime.h>

#define IN_  1024
#define H_   1024
#define NQ_  2
#define B_   128
#define T_   512
#define TH2  512           // 16 waves of 32
#define UPT  2             // hidden units per thread (1024 / 512)

typedef float v2f __attribute__((ext_vector_type(2)));
typedef float v8f __attribute__((ext_vector_type(8)));

// ---------------------------------------------------------------------------
// Phase 1: q_x for layer 0 over all (b,t) rows via V_WMMA_F32_16X16X4_F32.
//   u = W_xh1 @ x_row + b_xh1   (2 values, via 16x16x4 f32 WMMA, K-loop 256)
//   q = W_xh2 @ u     + b_xh2   (2x2 affine)
// One wave handles 16 rows. A: 16x4 tile of x. B: 4x16 with only N=0,1 live.
// ---------------------------------------------------------------------------
__global__ __launch_bounds__(32) void qx_phase_wmma(
    const float* __restrict__ x,        // [B*T, IN]
    const float* __restrict__ Wx1,      // [2, 1024]
    const float* __restrict__ bx1,      // [2]
    const float* __restrict__ Wx2,      // [2, 2]
    const float* __restrict__ bx2,      // [2]
    float* __restrict__ qx)             // [T, B, 2]
{
  __shared__ float u_sh[16][2];
  const int  lane = threadIdx.x;
  const int  m    = lane & 15;          // A: row / B: col / shared by layout
  const int  half = lane >> 4;          // K-half select
  const long r0   = (long)blockIdx.x * 16;
  const float* xrow = x + (r0 + m) * IN_;

  // B-matrix operand: clamp row index, mask dead columns (EXEC stays all-1s)
  const int   nc    = (m < NQ_) ? m : 0;
  const float nmask = (m < NQ_) ? 1.0f : 0.0f;
  const float* wrow = Wx1 + nc * IN_;

  v8f acc = {};
  for (int k = 0; k < IN_; k += 4) {
    v2f a, bm;
    // A 16x4 f32 layout: lane m (both halves), vgpr j holds K = 2*half + j
    a.x = xrow[k + 2 * half + 0];
    a.y = xrow[k + 2 * half + 1];
    // B 4x16 f32 layout mirrors A: col = m, K = 2*half + j
    bm.x = nmask * wrow[k + 2 * half + 0];
    bm.y = nmask * wrow[k + 2 * half + 1];
    acc = __builtin_amdgcn_wmma_f32_16x16x4_f32(false, a, false, bm,
                                                (short)0, acc, false, false);
  }

  // D 16x16 f32: lane (half*16 + n) vgpr j holds D[half*8 + j][n]
  if (m < NQ_) {
#pragma unroll
    for (int j = 0; j < 8; ++j)
      u_sh[half * 8 + j][m] = acc[j] + bx1[m];
  }
  __syncthreads();

  if (lane < 16) {
    const float u0 = u_sh[lane][0];
    const float u1 = u_sh[lane][1];
    const float q0 = fmaf(u0, Wx2[0], fmaf(u1, Wx2[1], bx2[0]));
    const float q1 = fmaf(u0, Wx2[2], fmaf(u1, Wx2[3], bx2[1]));
    const long r = r0 + lane;
    const int  bb = (int)(r >> 9);       // / T
    const int  t  = (int)(r & (T_ - 1)); // % T
    qx[((long)t * B_ + bb) * 2 + 0] = q0;
    qx[((long)t * B_ + bb) * 2 + 1] = q1;
  }
}

// ---------------------------------------------------------------------------
// Phase 2: sequential recurrence, one workgroup per batch element.
// ---------------------------------------------------------------------------
__device__ __forceinline__ float wred32(float v) {
  v += __shfl_xor(v, 16, 32);
  v += __shfl_xor(v, 8, 32);
  v += __shfl_xor(v, 4, 32);
  v += __shfl_xor(v, 2, 32);
  v += __shfl_xor(v, 1, 32);
  return v;
}
__device__ __forceinline__ float sigm(float x) {
  return 1.0f / (1.0f + __expf(-x));
}
__device__ __forceinline__ float tanh_f(float x) {
  const float e = __expf(2.0f * x);
  return (e - 1.0f) / (e + 1.0f);
}

// gate = cw0*a0 + cw1*a1 + cw2*a2 + cw3*a3 + cw4  (all indices constant-folded)
#define GATE(l, g, s, a0, a1, a2, a3)                                          \
  fmaf(cw[l][g][s][0], (a0),                                                   \
  fmaf(cw[l][g][s][1], (a1),                                                   \
  fmaf(cw[l][g][s][2], (a2),                                                   \
  fmaf(cw[l][g][s][3], (a3), cw[l][g][s][4]))))

__global__ __launch_bounds__(TH2) void recur_phase(
    const float* __restrict__ qxbuf,                                  // [T,B,2]
    const float* __restrict__ Wxh3_0, const float* __restrict__ bxh3_0,
    const float* __restrict__ Whh1_0, const float* __restrict__ bhh1_0,
    const float* __restrict__ Whh2_0, const float* __restrict__ bhh2_0,
    const float* __restrict__ Whh3_0, const float* __restrict__ bhh3_0,
    const float* __restrict__ Wxh1_1, const float* __restrict__ bxh1_1,
    const float* __restrict__ Wxh2_1, const float* __restrict__ bxh2_1,
    const float* __restrict__ Wxh3_1, const float* __restrict__ bxh3_1,
    const float* __restrict__ Whh1_1, const float* __restrict__ bhh1_1,
    const float* __restrict__ Whh2_1, const float* __restrict__ bhh2_1,
    const float* __restrict__ Whh3_1, const float* __restrict__ bhh3_1,
    float* __restrict__ out)       // outs[B,T,H] ++ hT[2,B,H] ++ cT[2,B,H]
{
  const int tid = threadIdx.x;
  const int b   = blockIdx.x;
  __shared__ float red0[16][2];   // layer-0 h-path partials per wave
  __shared__ float red1[16][4];   // layer-1 x-path + h-path partials per wave

  // --- register-resident weights (held for the whole T loop) ---
  float cw[2][4][UPT][5];               // expansion coeffs + fused bias
  float rh0[2][UPT], rx1[2][UPT], rh1[2][UPT];
#pragma unroll
  for (int s = 0; s < UPT; ++s) {
    const int j = tid + TH2 * s;
#pragma unroll
    for (int g = 0; g < 4; ++g) {
      const int row = g * H_ + j;
      cw[0][g][s][0] = Wxh3_0[row * 2 + 0];
      cw[0][g][s][1] = Wxh3_0[row * 2 + 1];
      cw[0][g][s][2] = Whh3_0[row * 2 + 0];
      cw[0][g][s][3] = Whh3_0[row * 2 + 1];
      cw[0][g][s][4] = bxh3_0[row] + bhh3_0[row];
      cw[1][g][s][0] = Wxh3_1[row * 2 + 0];
      cw[1][g][s][1] = Wxh3_1[row * 2 + 1];
      cw[1][g][s][2] = Whh3_1[row * 2 + 0];
      cw[1][g][s][3] = Whh3_1[row * 2 + 1];
      cw[1][g][s][4] = bxh3_1[row] + bhh3_1[row];
    }
#pragma unroll
    for (int d = 0; d < 2; ++d) {
      rh0[d][s] = Whh1_0[d * H_ + j];
      rx1[d][s] = Wxh1_1[d * H_ + j];
      rh1[d][s] = Whh1_1[d * H_ + j];
    }
  }
  // uniform scalars (compiler keeps these in SGPRs)
  const float bh1_0a = bhh1_0[0], bh1_0b = bhh1_0[1];
  const float Wh2_0[4] = {Whh2_0[0], Whh2_0[1], Whh2_0[2], Whh2_0[3]};
  const float bh2_0a = bhh2_0[0], bh2_0b = bhh2_0[1];
  const float bx1_1a = bxh1_1[0], bx1_1b = bxh1_1[1];
  const float Wx2_1[4] = {Wxh2_1[0], Wxh2_1[1], Wxh2_1[2], Wxh2_1[3]};
  const float bx2_1a = bxh2_1[0], bx2_1b = bxh2_1[1];
  const float bh1_1a = bhh1_1[0], bh1_1b = bhh1_1[1];
  const float Wh2_1[4] = {Whh2_1[0], Whh2_1[1], Whh2_1[2], Whh2_1[3]};
  const float bh2_1a = bhh2_1[0], bh2_1b = bhh2_1[1];

  float h0[UPT] = {0.f, 0.f}, c0[UPT] = {0.f, 0.f};
  float h1[UPT] = {0.f, 0.f}, c1[UPT] = {0.f, 0.f};
  const int  wave  = tid >> 5;
  const bool lane0 = (tid & 31) == 0;

  for (int t = 0; t < T_; ++t) {
    // precomputed layer-0 x-path bottleneck (uniform -> scalar load)
    const float qx0 = qxbuf[((long)t * B_ + b) * 2 + 0];
    const float qx1 = qxbuf[((long)t * B_ + b) * 2 + 1];

    // ---- layer 0: h-path rank-2 reduction ----
    float p0 = 0.f, p1 = 0.f;
#pragma unroll
    for (int s = 0; s < UPT; ++s) {
      p0 = fmaf(rh0[0][s], h0[s], p0);
      p1 = fmaf(rh0[1][s], h0[s], p1);
    }
    p0 = wred32(p0); p1 = wred32(p1);
    if (lane0) { red0[wave][0] = p0; red0[wave][1] = p1; }
    __syncthreads();
    float U0 = bh1_0a, U1 = bh1_0b;
#pragma unroll
    for (int w = 0; w < 16; ++w) { U0 += red0[w][0]; U1 += red0[w][1]; }
    const float qh0 = fmaf(U0, Wh2_0[0], fmaf(U1, Wh2_0[1], bh2_0a));
    const float qh1 = fmaf(U0, Wh2_0[2], fmaf(U1, Wh2_0[3], bh2_0b));

    // ---- layer 0: gates + cell update ----
#pragma unroll
    for (int s = 0; s < UPT; ++s) {
      const float i_t = sigm(GATE(0, 0, s, qx0, qx1, qh0, qh1));
      const float f_t = sigm(GATE(0, 1, s, qx0, qx1, qh0, qh1));
      const float g_t = tanh_f(GATE(0, 2, s, qx0, qx1, qh0, qh1));
      const float o_t = sigm(GATE(0, 3, s, qx0, qx1, qh0, qh1));
      c0[s] = fmaf(c0[s], f_t, i_t * g_t);
      h0[s] = o_t * tanh_f(c0[s]);
    }

    // ---- layer 1: fused x-path (on fresh h0) + h-path reductions ----
    float qa = 0.f, qb = 0.f, qc = 0.f, qd = 0.f;
#pragma unroll
    for (int s = 0; s < UPT; ++s) {
      qa = fmaf(rx1[0][s], h0[s], qa);
      qb = fmaf(rx1[1][s], h0[s], qb);
      qc = fmaf(rh1[0][s], h1[s], qc);
      qd = fmaf(rh1[1][s], h1[s], qd);
    }
    qa = wred32(qa); qb = wred32(qb); qc = wred32(qc); qd = wred32(qd);
    if (lane0) {
      red1[wave][0] = qa; red1[wave][1] = qb;
      red1[wave][2] = qc; red1[wave][3] = qd;
    }
    __syncthreads();
    float X0 = bx1_1a, X1 = bx1_1b, V0 = bh1_1a, V1 = bh1_1b;
#pragma unroll
    for (int w = 0; w < 16; ++w) {
      X0 += red1[w][0]; X1 += red1[w][1];
      V0 += red1[w][2]; V1 += red1[w][3];
    }
    const float qxa = fmaf(X0, Wx2_1[0], fmaf(X1, Wx2_1[1], bx2_1a));
    const float qxb = fmaf(X0, Wx2_1[2], fmaf(X1, Wx2_1[3], bx2_1b));
    const float qha = fmaf(V0, Wh2_1[0], fmaf(V1, Wh2_1[1], bh2_1a));
    const float qhb = fmaf(V0, Wh2_1[2], fmaf(V1, Wh2_1[3], bh2_1b));

    // ---- layer 1: gates + cell update + output ----
#pragma unroll
    for (int s = 0; s < UPT; ++s) {
      const float i_t = sigm(GATE(1, 0, s, qxa, qxb, qha, qhb));
      const float f_t = sigm(GATE(1, 1, s, qxa, qxb, qha, qhb));
      const float g_t = tanh_f(GATE(1, 2, s, qxa, qxb, qha, qhb));
      const float o_t = sigm(GATE(1, 3, s, qxa, qxb, qha, qhb));
      c1[s] = fmaf(c1[s], f_t, i_t * g_t);
      h1[s] = o_t * tanh_f(c1[s]);
      out[((long)b * T_ + t) * H_ + tid + TH2 * s] = h1[s];
    }
  }

  // final states: hT then cT, each [L,B,H]
  const long OUT0 = (long)B_ * T_ * H_;
#pragma unroll
  for (int s = 0; s < UPT; ++s) {
    const int j = tid + TH2 * s;
    out[OUT0 + ((long)0 * B_ + b) * H_ + j] = h0[s];
    out[OUT0 + ((long)1 * B_ + b) * H_ + j] = h1[s];
    out[OUT0 + (long)2 * B_ * H_ + ((long)0 * B_ + b) * H_ + j] = c0[s];
    out[OUT0 + (long)2 * B_ * H_ + ((long)1 * B_ + b) * H_ + j] = c1[s];
  }
}

// ---------------------------------------------------------------------------
extern "C" void kernel_launch(void* const* d_in, const int* in_sizes, int n_in,
                              void* d_out, int out_size, void* d_ws,
                              size_t ws_size, hipStream_t stream)
{
  (void)in_sizes; (void)n_in; (void)out_size; (void)ws_size;
  const float* x = (const float*)d_in[0];
  // per layer (12 tensors): W_xh1,b_xh1,W_xh2,b_xh2,W_xh3,b_xh3,
  //                         W_hh1,b_hh1,W_hh2,b_hh2,W_hh3,b_hh3
  const float* P[24];
  for (int i = 0; i < 24; ++i) P[i] = (const float*)d_in[1 + i];

  float* qx  = (float*)d_ws;   // [T, B, 2] = 512 KB
  float* out = (float*)d_out;

  qx_phase_wmma<<<dim3((B_ * T_) / 16), dim3(32), 0, stream>>>(
      x, P[0], P[1], P[2], P[3], qx);

  recur_phase<<<dim3(B_), dim3(TH2), 0, stream>>>(
      qx,
      P[4], P[5], P[6], P[7], P[8], P[9], P[10], P[11],
      P[12], P[13], P[14], P[15], P[16], P[17],
      P[18], P[19], P[20], P[21], P[22], P[23],
      out);
}